// irnn_layer_58265526337601
// MI455X (gfx1250) — compile-verified
//
#include <hip/hip_runtime.h>
#include <hip/hip_bf16.h>

// IRNN 8-direction spatial scan, B=8, C=32, H=256, W=256, f32.
// d_out = concat(up, right, down, left, zuoxia, youxia, zuoshang, youshang),
// each 16,777,216 floats. Weights: up->w_up, right->w_right, left->w_left,
// down + all four diagonals->w_down (faithful to reference).

#define IRNN_H 256
#define IRNN_W 256
#define IRNN_PLANES 256          // B*C
#define IRNN_PLANE  (IRNN_H * IRNN_W)
#define TILE_P 33                // padded LDS row stride (odd => conflict-free column reads)

// ---- optional CDNA5 async global->LDS path (guarded; clean fallback) -------
#if defined(__has_builtin)
#if __has_builtin(__builtin_amdgcn_global_load_async_to_lds_b32)
#define IRNN_ASYNC 1
#endif
#endif

#if defined(IRNN_ASYNC)
typedef __attribute__((address_space(3))) int lds_int;
__device__ __forceinline__ lds_int* lds_cast(void* p) {
    // low 32 bits of the flat LDS address are the LDS byte offset
    return (lds_int*)(uint32_t)(uintptr_t)p;
}
__device__ __forceinline__ int* gbl_cast(const void* p) {
    return (int*)(uintptr_t)p;
}
__device__ __forceinline__ void wait_async_le1() {
#if __has_builtin(__builtin_amdgcn_s_wait_asynccnt)
    __builtin_amdgcn_s_wait_asynccnt(1);
#else
    asm volatile("s_wait_asynccnt 0x1" ::: "memory");
#endif
}
__device__ __forceinline__ void wait_async_le0() {
#if __has_builtin(__builtin_amdgcn_s_wait_asynccnt)
    __builtin_amdgcn_s_wait_asynccnt(0);
#else
    asm volatile("s_wait_asynccnt 0x0" ::: "memory");
#endif
}
#endif

__device__ __forceinline__ float relu_fma(float carry, float w, float xv) {
    return fmaxf(fmaf(carry, w, xv), 0.0f);
}

// Cooperative, fully coalesced stage of a 256x32 column tile into padded LDS.
// Per wave instruction: 4 rows x 32 cols = 4 fully covered 128B lines.
__device__ __forceinline__ void stage_tile(const float* __restrict__ xpl,
                                           float* __restrict__ xt, int w0, int t) {
    #pragma unroll
    for (int s = 0; s < 8; ++s) {
        const int hh = s * 32 + (t >> 3);
        const int cc = (t & 7) * 4;
        const float4 v = *(const float4*)(xpl + (size_t)hh * IRNN_W + w0 + cc);
        float* dst = xt + hh * TILE_P + cc;
        dst[0] = v.x; dst[1] = v.y; dst[2] = v.z; dst[3] = v.w;
    }
}

__global__ __launch_bounds__(256) void irnn_scan_kernel(
    const float* __restrict__ x,
    const float* __restrict__ w_up_p,
    const float* __restrict__ w_down_p,
    const float* __restrict__ w_left_p,
    const float* __restrict__ w_right_p,
    float* __restrict__ o_up,    float* __restrict__ o_right,
    float* __restrict__ o_down,  float* __restrict__ o_left,
    float* __restrict__ o_zx,    float* __restrict__ o_yx,
    float* __restrict__ o_zs,    float* __restrict__ o_ys)
{
    constexpr int H = IRNN_H, W = IRNN_W;

    __shared__ float zb[2][W];               // diagonal carry exchange (double buffered)
    __shared__ float yb[2][W];
    __shared__ union {
        float xrow[2][W];                    // row path: async-staged input rows
        float xt[IRNN_H * TILE_P];           // col path: 256x32 padded input tile (~33 KB)
    } sx;

    const int t = threadIdx.x;

    if (blockIdx.x < IRNN_PLANES) {
        // ============== Row-direction scans: down, zuoxia, youxia, up ======
        const int plane = blockIdx.x;
        const size_t pb = (size_t)plane * IRNN_PLANE;
        const float wd = w_down_p[0];
        const float wu = w_up_p[0];
        const float* __restrict__ xp = x + pb;
        float* __restrict__ od = o_down + pb;
        float* __restrict__ oz = o_zx  + pb;
        float* __restrict__ oy = o_yx  + pb;
        float* __restrict__ ou = o_up  + pb;
        const int k = t;

        float dcar = 0.0f;
#if defined(IRNN_ASYNC)
        // Software pipeline: row h+1 streams into LDS via the async engine
        // (ASYNCcnt) while row h is consumed. Loads retire in order, so
        // asynccnt<=1 guarantees the older row has landed.
        __builtin_amdgcn_global_load_async_to_lds_b32(
            gbl_cast(xp + k), lds_cast(&sx.xrow[0][k]), 0, 0);
#endif
        for (int h = 0; h < H; ++h) {
#if defined(IRNN_ASYNC)
            if (h + 1 < H) {
                __builtin_amdgcn_global_load_async_to_lds_b32(
                    gbl_cast(xp + (size_t)(h + 1) * W + k),
                    lds_cast(&sx.xrow[(h + 1) & 1][k]), 0, 0);
                wait_async_le1();
            } else {
                wait_async_le0();
            }
            asm volatile("" ::: "memory");
            const float xv = sx.xrow[h & 1][k];
#else
            const float xv = xp[h * W + k];
            if (h + 8 < H) __builtin_prefetch(xp + (h + 8) * W + k, 0, 3);
#endif
            float d, z, y;
            if (h == 0) {
                d = xv; z = xv; y = xv;      // first line pass-through
            } else {
                d = relu_fma(dcar, wd, xv);
                const float zp = zb[(h - 1) & 1][(k == 0) ? 0 : (k - 1)];
                z = (k == 0) ? xv : relu_fma(zp, wd, xv);
                const float yp = yb[(h - 1) & 1][(k == W - 1) ? (W - 1) : (k + 1)];
                y = (k == W - 1) ? xv : relu_fma(yp, wd, xv);
            }
            __builtin_nontemporal_store(d, od + h * W + k);
            __builtin_nontemporal_store(z, oz + h * W + k);
            __builtin_nontemporal_store(y, oy + h * W + k);
            dcar = d;
            zb[h & 1][k] = z;
            yb[h & 1][k] = y;
            __syncthreads();                 // publish line h for the diagonals
        }

        // Backward over rows (up scan); x re-read hits L2.
        float ucar = 0.0f;
        for (int h = H - 1; h >= 0; --h) {
            const float xv = xp[h * W + k];
            if (h >= 8) __builtin_prefetch(xp + (h - 8) * W + k, 0, 3);
            const float u = (h == H - 1) ? xv : relu_fma(ucar, wu, xv);
            __builtin_nontemporal_store(u, ou + h * W + k);
            ucar = u;
        }
    } else {
        // ============== Column-direction scans: right, zuoshang, youshang, left
        const int plane = blockIdx.x - IRNN_PLANES;
        const size_t pb = (size_t)plane * IRNN_PLANE;
        const float wd = w_down_p[0];
        const float wr = w_right_p[0];
        const float wl = w_left_p[0];
        const int h = t;                     // this thread owns image row h
        const float* __restrict__ xpl = x + pb;
        float* __restrict__ orp = o_right + pb + (size_t)h * W;
        float* __restrict__ olp = o_left  + pb + (size_t)h * W;
        float* __restrict__ ozp = o_zs    + pb + (size_t)h * W;
        float* __restrict__ oyp = o_ys    + pb + (size_t)h * W;
        float* xt = sx.xt;

        // Forward over columns in 32-wide tiles (right + both column diagonals).
        float rcar = 0.0f;
        for (int w0 = 0; w0 < W; w0 += 32) {
            stage_tile(xpl, xt, w0, t);      // coalesced global -> LDS
            __syncthreads();
            for (int w4 = w0; w4 < w0 + 32; w4 += 4) {
                float rr[4], zz[4], yy[4];
                #pragma unroll
                for (int j = 0; j < 4; ++j) {
                    const int w = w4 + j;
                    const float xv = xt[h * TILE_P + (w - w0)];  // conflict-free (odd stride)
                    float r, z, y;
                    if (w == 0) {
                        r = xv; z = xv; y = xv;   // first column pass-through
                    } else {
                        r = relu_fma(rcar, wr, xv);
                        const float zp = zb[(w - 1) & 1][(h == 0) ? 0 : (h - 1)];
                        z = (h == 0) ? xv : relu_fma(zp, wd, xv);
                        const float yp = yb[(w - 1) & 1][(h == H - 1) ? (H - 1) : (h + 1)];
                        y = (h == H - 1) ? xv : relu_fma(yp, wd, xv);
                    }
                    rcar = r; rr[j] = r; zz[j] = z; yy[j] = y;
                    zb[w & 1][h] = z;
                    yb[w & 1][h] = y;
                    __syncthreads();         // per-column publish for diagonals
                }
                // Per-thread contiguous stores; L2 write-combines into full lines.
                *(float4*)(orp + w4) = make_float4(rr[0], rr[1], rr[2], rr[3]);
                *(float4*)(ozp + w4) = make_float4(zz[0], zz[1], zz[2], zz[3]);
                *(float4*)(oyp + w4) = make_float4(yy[0], yy[1], yy[2], yy[3]);
            }
            // last per-step barrier already fences xt reuse for the next tile
        }

        // Backward over columns (left scan): no per-step barrier needed.
        float lcar = 0.0f;
        for (int w0 = W - 32; w0 >= 0; w0 -= 32) {
            stage_tile(xpl, xt, w0, t);
            __syncthreads();
            for (int w4 = w0 + 28; w4 >= w0; w4 -= 4) {
                float ll[4];
                #pragma unroll
                for (int j = 3; j >= 0; --j) {
                    const int w = w4 + j;
                    const float xv = xt[h * TILE_P + (w - w0)];
                    const float l = (w == W - 1) ? xv : relu_fma(lcar, wl, xv);
                    lcar = l; ll[j] = l;
                }
                *(float4*)(olp + w4) = make_float4(ll[0], ll[1], ll[2], ll[3]);
            }
            __syncthreads();                 // all tile reads done before restage
        }
    }
}

extern "C" void kernel_launch(void* const* d_in, const int* in_sizes, int n_in,
                              void* d_out, int out_size, void* d_ws, size_t ws_size,
                              hipStream_t stream) {
    (void)in_sizes; (void)n_in; (void)out_size; (void)d_ws; (void)ws_size;

    const float* x       = (const float*)d_in[0];
    const float* w_up    = (const float*)d_in[1];
    const float* w_down  = (const float*)d_in[2];
    const float* w_left  = (const float*)d_in[3];
    const float* w_right = (const float*)d_in[4];

    float* out = (float*)d_out;
    const size_t OS = (size_t)IRNN_PLANES * IRNN_PLANE;   // 16,777,216 per output

    irnn_scan_kernel<<<dim3(2 * IRNN_PLANES), dim3(256), 0, stream>>>(
        x, w_up, w_down, w_left, w_right,
        out + 0 * OS,   // top_up
        out + 1 * OS,   // top_right
        out + 2 * OS,   // top_down
        out + 3 * OS,   // top_left
        out + 4 * OS,   // top_zuoxia
        out + 5 * OS,   // top_youxia
        out + 6 * OS,   // top_zuoshang
        out + 7 * OS);  // top_youshang
}